// Intt_58815282152154
// MI455X (gfx1250) — compile-verified
//
#include <hip/hip_runtime.h>
#include <stdint.h>
#include <stddef.h>

// ---------------------------------------------------------------------------
// Inverse NTT over F_P, P = 2013265921 = 15*2^27 + 1, N = 2^22, C = 4 columns.
// Exact u32 arithmetic, Montgomery multiplication (R = 2^32).
// Two LDS-tiled passes (11 stages each); the 64 MB working set is L2-resident.
// CDNA5 paths: async global->LDS DMA gather (ASYNCcnt + s_wait_asynccnt) in
// phase 1; Tensor Data Mover strided-tile load with LDS padding (TENSORcnt +
// s_wait_tensorcnt) in phase 2; global_prefetch_b8 in the fallback path.
// ---------------------------------------------------------------------------

#define LOGN     22
#define NTT_N    (1u << LOGN)
#define HALF_N   (1u << (LOGN - 1))
#define TILE_LOG 11
#define TILE     (1u << TILE_LOG)     // 2048 points per tile
#define NCOLS    4
#define THREADS  256
#define ROWP     5                    // padded LDS row pitch (DWORDs) via TDM pad

static constexpr uint32_t MODP = 2013265921u;   // 2^31 - 2^27 + 1

__host__ __device__ constexpr uint32_t calc_neg_pinv() {
  uint32_t inv = 1u;                            // Newton: inv = P^-1 mod 2^32
  for (int i = 0; i < 5; ++i) inv *= 2u - MODP * inv;
  return 0u - inv;                              // -P^-1 mod 2^32
}
__host__ __device__ constexpr uint32_t calc_r2() {
  unsigned long long r = (1ull << 32) % MODP;   // 2^32 mod P
  return (uint32_t)((r * r) % MODP);            // 2^64 mod P
}
static constexpr uint32_t PINV_NEG = calc_neg_pinv();
static constexpr uint32_t R2MODP   = calc_r2();

__device__ __forceinline__ uint32_t montmul(uint32_t a, uint32_t b) {
  uint64_t t = (uint64_t)a * (uint64_t)b;
  uint32_t m = (uint32_t)t * PINV_NEG;
  uint32_t r = (uint32_t)((t + (uint64_t)m * (uint64_t)MODP) >> 32);
  return (r >= MODP) ? r - MODP : r;
}
__device__ __forceinline__ uint32_t addmod(uint32_t a, uint32_t b) {
  uint32_t s = a + b;  return (s >= MODP) ? s - MODP : s;
}
__device__ __forceinline__ uint32_t submod(uint32_t a, uint32_t b) {
  uint32_t s = a + MODP - b;  return (s >= MODP) ? s - MODP : s;
}
__device__ __forceinline__ uint32_t brev_tile(uint32_t v) {
  return __brev(v) >> (32 - TILE_LOG);          // 11-bit bit reversal
}

// --------------------- CDNA5 async global -> LDS DMA -----------------------
#if defined(__has_builtin)
#  if __has_builtin(__builtin_amdgcn_global_load_async_to_lds_b32)
#    define HAVE_ASYNC_G2L 1
#  endif
#  if __has_builtin(__builtin_amdgcn_tensor_load_to_lds)
#    define HAVE_TDM 1
#  endif
#endif
#ifndef HAVE_ASYNC_G2L
#  define HAVE_ASYNC_G2L 0
#endif
#ifndef HAVE_TDM
#  define HAVE_TDM 0
#endif

// clang (HIP mode) prints addrspace(1) as '__device__', addrspace(3) as
// '__shared__'; these builtins take addrspace-qualified int*.
typedef __attribute__((address_space(1))) int as1_int;
typedef __attribute__((address_space(3))) int as3_int;

__device__ __forceinline__ void async_g2l_b32(const uint32_t* g, uint32_t* l) {
#if HAVE_ASYNC_G2L
  __builtin_amdgcn_global_load_async_to_lds_b32(
      (as1_int*)g, (as3_int*)l, /*imm offset*/ 0, /*cpol*/ 0);
#else
  *l = *g;
#endif
}

__device__ __forceinline__ void wait_async_lds() {
#if HAVE_ASYNC_G2L
#  if __has_builtin(__builtin_amdgcn_s_wait_asynccnt)
  __builtin_amdgcn_s_wait_asynccnt(0);
#  else
  asm volatile("s_wait_asynccnt 0" ::: "memory");
#  endif
#endif
}

#if HAVE_TDM
typedef unsigned int v4u __attribute__((ext_vector_type(4)));
typedef int          v4i __attribute__((ext_vector_type(4)));
typedef int          v8i __attribute__((ext_vector_type(8)));

// One TDM descriptor per workgroup: 2-D tile, 2048 rows x 16 B, row stride
// 32 KB, loaded into LDS with 1 DWORD of padding after every 4 DWORDs
// (pad_interval code 1 = 4 DWORDs, pad_amount code 0 = 1 DWORD) -> LDS row
// pitch of 5 DWORDs, bank-conflict-free for all butterfly strides.
__device__ __forceinline__ void tdm_load_comb(const uint32_t* gbase,
                                              uint32_t lds_byte_addr) {
  uint64_t ga = (uint64_t)(uintptr_t)gbase;     // tile start (57-bit address)

  v4u g0;
  g0.x = 1u;                                    // count=1, load, no gather
  g0.y = lds_byte_addr;                         // lds_addr  (bits  63:32)
  g0.z = (uint32_t)ga;                          // global_addr[31:0] (95:64)
  g0.w = ((uint32_t)(ga >> 32) & 0x01FFFFFFu)   // global_addr[56:32] (120:96)
       | (2u << 30);                            // type=2 ("image")

  v8i g1;
  g1[0] = (int)((2u << 16)                      // data_size = 4 bytes
              | (1u << 20)                      // pad_enable
              | (1u << 22));                    // pad_interval = 4 DWORDs
                                                // pad_amount  = 1 DWORD (code 0)
  g1[1] = (int)(4u << 16);                      // tensor_dim0 = 4 (low 16 -> [31:16])
  g1[2] = (int)(TILE << 16);                    // tensor_dim0 hi=0 | tensor_dim1 lo = 2048
  g1[3] = (int)(4u << 16);                      // tensor_dim1 hi=0 | tile_dim0 = 4
  g1[4] = (int)TILE;                            // tile_dim1 = 2048, tile_dim2 = 0
  g1[5] = (int)(TILE * NCOLS);                  // tensor_dim0_stride = 8192 u32
  g1[6] = 0;                                    // stride hi, dim1_stride lo
  g1[7] = 0;

  v4i z4 = {0, 0, 0, 0};                        // groups 2/3 unused (2-D tensor)
  v8i z8 = {0, 0, 0, 0, 0, 0, 0, 0};            // extra group (clang-23 form)
  __builtin_amdgcn_tensor_load_to_lds(g0, g1, z4, z4, z8, /*cpol*/ 0);
}
#endif

// ------------------- kernel 1: twiddles -> Montgomery domain ----------------
__global__ __launch_bounds__(THREADS)
void twiddle_prep(const uint32_t* __restrict__ tw,
                  const uint32_t* __restrict__ ninv,
                  uint32_t* __restrict__ wsm) {
  uint32_t i = blockIdx.x * THREADS + threadIdx.x;
  if (i < HALF_N) wsm[i] = montmul(tw[i], R2MODP);        // w*R mod P
  if (i == 0)     wsm[HALF_N] = montmul(ninv[0], R2MODP); // n_inv*R mod P
}

// ------------- kernel 2: bit-reversal gather + stages 1..11 ----------------
__global__ __launch_bounds__(THREADS)
void intt_phase1(const uint32_t* __restrict__ in,
                 uint32_t* __restrict__ out,
                 const uint32_t* __restrict__ tws,
                 int tw_is_mont) {
  __shared__ uint32_t lds[NCOLS * TILE];                  // 32 KB, SoA by column
  const uint32_t tid = threadIdx.x;
  const uint32_t b   = blockIdx.x;                        // tile id 0..2047
  const uint32_t c0  = brev_tile(b);

  // lds[col][j] = x[rev22(b*TILE + j)][col] = x[brev11(j)*TILE + c0][col].
  // Per-lane async DMA absorbs bit reversal + AoS->SoA into gather addresses.
  #pragma unroll
  for (uint32_t it = 0; it < TILE / THREADS; ++it) {
    uint32_t j = tid + it * THREADS;
    uint32_t k = brev_tile(j);
    const uint32_t* src = in + ((size_t)k * TILE + c0) * NCOLS;
    #pragma unroll
    for (uint32_t col = 0; col < NCOLS; ++col)
      async_g2l_b32(src + col, &lds[col * TILE + j]);
  }
  wait_async_lds();
  __syncthreads();

  // Stages s = 1..11 (m = 2..2048) entirely inside the tile.
  for (uint32_t s = 1; s <= TILE_LOG; ++s) {
    const uint32_t half  = 1u << (s - 1);
    const uint32_t shift = LOGN - s;                      // tw index = j << shift
    for (uint32_t q = tid; q < NCOLS * (TILE / 2); q += THREADS) {
      const uint32_t col  = q >> (TILE_LOG - 1);
      const uint32_t bq   = q & (TILE / 2 - 1);
      const uint32_t blk  = bq >> (s - 1);
      const uint32_t j    = bq & (half - 1);
      const uint32_t i0   = (blk << s) + j;
      const uint32_t i1   = i0 + half;
      const uint32_t base = col * TILE;
      uint32_t lo = lds[base + i0];
      uint32_t hi = lds[base + i1];
      uint32_t w  = tws[j << shift];
      if (!tw_is_mont) w = montmul(w, R2MODP);
      uint32_t t = montmul(hi, w);
      lds[base + i0] = addmod(lo, t);
      lds[base + i1] = submod(lo, t);
    }
    __syncthreads();
  }

  // Natural-order contiguous write-back, 16 B per point (4 columns).
  #pragma unroll
  for (uint32_t it = 0; it < TILE / THREADS; ++it) {
    uint32_t i  = tid + it * THREADS;
    size_t   gi = (size_t)b * TILE + i;
    uint4 v;
    v.x = lds[0 * TILE + i];
    v.y = lds[1 * TILE + i];
    v.z = lds[2 * TILE + i];
    v.w = lds[3 * TILE + i];
    ((uint4*)out)[gi] = v;
  }
}

// ------- kernel 3: stages 12..22 on stride-2048 combs + n^-1 scaling -------
__global__ __launch_bounds__(THREADS)
void intt_phase2(uint32_t* __restrict__ data,
                 const uint32_t* __restrict__ tws,
                 int tw_is_mont,
                 const uint32_t* __restrict__ ninv_src,
                 int ninv_is_mont) {
  // Padded AoS layout written by the TDM: element (mp, col) at lds2[mp*5+col].
  __shared__ uint32_t lds2[TILE * ROWP + NCOLS];          // ~40 KB
  const uint32_t tid = threadIdx.x;
  const uint32_t g   = blockIdx.x;                        // low-11-bit class
  uint4* data4 = (uint4*)data;

  uint32_t ninv_m = ninv_src[0];
  if (!ninv_is_mont) ninv_m = montmul(ninv_m, R2MODP);

#if HAVE_TDM
  // One descriptor gathers the whole comb {i : i mod TILE == g}:
  // 2048 rows x 16 B at 32 KB stride, padded into stride-5-DWORD LDS rows.
  if (tid < 32) {                               // one wave issues the DMA
    tdm_load_comb(data + (size_t)g * NCOLS,
                  (uint32_t)(uintptr_t)(as3_int*)&lds2[0]);
    __builtin_amdgcn_s_wait_tensorcnt(0);
  }
  __syncthreads();
#else
  #pragma unroll
  for (uint32_t it = 0; it < TILE / THREADS; ++it) {
    uint32_t mp = tid + it * THREADS;
    size_t   gi = (size_t)mp * TILE + g;
    if (it + 1 < TILE / THREADS)
      __builtin_prefetch((const void*)(data4 + ((size_t)(mp + THREADS) * TILE + g)), 0, 0);
    uint4 v = data4[gi];
    lds2[mp * ROWP + 0] = v.x;
    lds2[mp * ROWP + 1] = v.y;
    lds2[mp * ROWP + 2] = v.z;
    lds2[mp * ROWP + 3] = v.w;
  }
  __syncthreads();
#endif

  // Local stage l == global stage s = l + 11.
  // Global butterfly position j = jloc*TILE + g; tw index = j << (11 - l).
  for (uint32_t l = 1; l <= TILE_LOG; ++l) {
    const uint32_t half  = 1u << (l - 1);
    const uint32_t shift = TILE_LOG - l;
    for (uint32_t q = tid; q < NCOLS * (TILE / 2); q += THREADS) {
      const uint32_t col = q >> (TILE_LOG - 1);
      const uint32_t bq  = q & (TILE / 2 - 1);
      const uint32_t blk = bq >> (l - 1);
      const uint32_t j   = bq & (half - 1);
      const uint32_t i0  = (blk << l) + j;
      const uint32_t i1  = i0 + half;
      uint32_t lo = lds2[i0 * ROWP + col];
      uint32_t hi = lds2[i1 * ROWP + col];
      uint32_t w  = tws[((j << TILE_LOG) + g) << shift];
      if (!tw_is_mont) w = montmul(w, R2MODP);
      uint32_t t = montmul(hi, w);
      lds2[i0 * ROWP + col] = addmod(lo, t);
      lds2[i1 * ROWP + col] = submod(lo, t);
    }
    __syncthreads();
  }

  // Scale by n^-1 (Montgomery form => plain product mod P) and scatter back.
  #pragma unroll
  for (uint32_t it = 0; it < TILE / THREADS; ++it) {
    uint32_t mp = tid + it * THREADS;
    size_t   gi = (size_t)mp * TILE + g;
    uint4 v;
    v.x = montmul(lds2[mp * ROWP + 0], ninv_m);
    v.y = montmul(lds2[mp * ROWP + 1], ninv_m);
    v.z = montmul(lds2[mp * ROWP + 2], ninv_m);
    v.w = montmul(lds2[mp * ROWP + 3], ninv_m);
    data4[gi] = v;
  }
}

// ---------------------------------------------------------------------------
extern "C" void kernel_launch(void* const* d_in, const int* in_sizes, int n_in,
                              void* d_out, int out_size, void* d_ws, size_t ws_size,
                              hipStream_t stream) {
  (void)in_sizes; (void)n_in; (void)out_size;
  // Inputs per setup_inputs(): [0] x (N*C values in [0,P)), [1] twiddles (N/2),
  // [2] n_inv (1 element). All values < P < 2^31, so 32-bit views are lossless.
  const uint32_t* x    = (const uint32_t*)d_in[0];
  const uint32_t* tw   = (const uint32_t*)d_in[1];
  const uint32_t* ninv = (const uint32_t*)d_in[2];
  uint32_t* out = (uint32_t*)d_out;
  uint32_t* wsm = (uint32_t*)d_ws;

  const size_t ws_needed = ((size_t)HALF_N + 1) * sizeof(uint32_t); // ~8 MB
  const bool use_ws = (d_ws != nullptr) && (ws_size >= ws_needed);

  if (use_ws)
    twiddle_prep<<<HALF_N / THREADS, THREADS, 0, stream>>>(tw, ninv, wsm);

  const uint32_t* tws  = use_ws ? wsm : tw;
  const uint32_t* nsrc = use_ws ? (wsm + HALF_N) : ninv;
  const int is_mont    = use_ws ? 1 : 0;

  intt_phase1<<<NTT_N / TILE, THREADS, 0, stream>>>(x, out, tws, is_mont);
  intt_phase2<<<NTT_N / TILE, THREADS, 0, stream>>>(out, tws, is_mont, nsrc, is_mont);
}